// KNNGraph_52785148068304
// MI455X (gfx1250) — compile-verified
//
#include <hip/hip_runtime.h>
#include <hip/hip_bf16.h>

// KNN graph via V_WMMA_F32_16X16X4_F32 on gfx1250 (wave32).
//
// key(m,n) = |p_m|^2 - 2 p_m . p_n  ==  d2(m,n) - |p_n|^2  (constant per center n)
// => top-k by key == top-k by d2. One WMMA per 16x16 (candidate x center) tile:
//   A (16x4, candidates): row m = (x, y, z, |p|^2)
//   B (4x16, centers):    col n = (-2x, -2y, -2z, 1)
//
// Selection state is u64-packed (monotone(key) << 32 | idx) so the lexicographic
// comparator (ascending key, then ascending index == jax top_k tie-break) is a
// single v_cmp_lt_u64 and all compare-exchanges are branchless cndmasks.
// A-operands are fetched as one ds_load_b64 per lane (components are adjacent in
// the float4 record) and software-pipelined one tile ahead of the WMMA.

#define NGRAPH   16
#define NPG      2048
#define KNN      16
#define MIN_DIST 5
#define EPSF     1e-10f

typedef __attribute__((ext_vector_type(2))) float v2f;
typedef __attribute__((ext_vector_type(8))) float v8f;

__device__ __forceinline__ unsigned int fmono(float f) {
    // order-preserving float -> u32 (handles negatives); fuses to v_bitop3
    unsigned int u = __float_as_uint(f);
    unsigned int mask = (unsigned int)((int)u >> 31) | 0x80000000u;
    return u ^ mask;
}

__global__ __launch_bounds__(256) void knn_wmma_kernel(
        const float* __restrict__ pos,      // [NGRAPH*NPG, 3]
        const int*   __restrict__ res_idx,  // [NGRAPH*NPG]
        int*         __restrict__ out)      // [NGRAPH*NPG*KNN, 2]
{
    __shared__ float4 sCand[NPG];           // 32 KB: (x, y, z, |p|^2) per node

    const int blk        = blockIdx.x;      // 256 blocks = 16 graphs x 16
    const int graph      = blk >> 4;
    const int blkInGraph = blk & 15;
    const int tid        = threadIdx.x;     // 0..255 (8 waves)
    const int wave       = tid >> 5;
    const int lane       = tid & 31;
    const int col        = lane & 15;
    const bool hi        = lane >= 16;
    const int base       = graph * NPG;

    // ---- stage the whole graph's candidates into LDS (cooperative) ----
    for (int c = tid; c < NPG; c += 256) {
        const float x = pos[(base + c) * 3 + 0];
        const float y = pos[(base + c) * 3 + 1];
        const float z = pos[(base + c) * 3 + 2];
        sCand[c] = make_float4(x, y, z, x * x + y * y + z * z);
    }
    __syncthreads();

    // lo lanes read (x,y) = half 0; hi lanes read (z,|p|^2) = half 1 -> one b64
    const float2* sC2 = (const float2*)sCand;
    const int comp = hi ? 1 : 0;

    // ---- this wave's 16 centers; build B operand once ----
    const int n0   = (blkInGraph * 8 + wave) * 16;  // local center tile base
    const int nloc = n0 + col;                      // this lane's center (local)
    const float2 bv = sC2[nloc * 2 + comp];
    v2f Bm;
    Bm.x = -2.0f * bv.x;                            // -2x (lo) / -2z (hi)
    Bm.y = hi ? 1.0f : -2.0f * bv.y;                // -2y (lo) /  1  (hi)

    // ---- per-lane top-16, ascending u64 list (slot 0 best, 15 worst) ----
    unsigned long long list[KNN];
#pragma unroll
    for (int j = 0; j < KNN; ++j) list[j] = ~0ull;

    const int rofs = hi ? 8 : 0;

    // ---- stream candidate tiles, A fetch pipelined one tile ahead ----
    float2 a_cur = sC2[col * 2 + comp];
    for (int m0 = 0; m0 < NPG; m0 += 16) {
        v2f Am;
        Am.x = a_cur.x;
        Am.y = a_cur.y;

        v8f acc = {};
        acc = __builtin_amdgcn_wmma_f32_16x16x4_f32(
                false, Am, false, Bm, (short)0, acc, false, false);

        // prefetch next tile's A while selection math runs (in-bounds clamp)
        const int mnext = (m0 + 16 < NPG) ? (m0 + 16) : m0;
        a_cur = sC2[(mnext + col) * 2 + comp];

        // pack all 8 candidates branchlessly; sentinel for self-loop
        unsigned long long pk[8];
        bool any = false;
#pragma unroll
        for (int r = 0; r < 8; ++r) {
            const int mloc = m0 + r + rofs;
            unsigned long long p =
                ((unsigned long long)fmono(acc[r]) << 32) | (unsigned int)mloc;
            p = (mloc == nloc) ? ~0ull : p;      // self -> never inserted
            pk[r] = p;
            any |= (p < list[KNN - 1]);
        }

        if (any) {                               // rare after warm-up
#pragma unroll
            for (int r = 0; r < 8; ++r) {
                if (pk[r] < list[KNN - 1]) {
                    list[KNN - 1] = pk[r];
#pragma unroll
                    for (int j = KNN - 1; j > 0; --j) {  // one bubble pass
                        const unsigned long long a = list[j - 1];
                        const unsigned long long b = list[j];
                        const bool sw = b < a;
                        list[j - 1] = sw ? b : a;
                        list[j]     = sw ? a : b;
                    }
                }
            }
        }
    }

    // ---- merge lane n (m%16<8 stream) with lane n+16 (m%16>=8 stream) ----
    unsigned long long other[KNN];
#pragma unroll
    for (int j = 0; j < KNN; ++j)
        other[j] = __shfl_xor(list[j], 16);

    // 16 smallest of two ascending 16-lists: elementwise min(a[j], b[15-j])
    unsigned long long c[KNN];
#pragma unroll
    for (int j = 0; j < KNN; ++j) {
        const unsigned long long a = list[j];
        const unsigned long long b = other[KNN - 1 - j];
        c[j] = (a < b) ? a : b;
    }
    // resulting sequence is bitonic: 4-stage bitonic clean-up sort (ascending)
#pragma unroll
    for (int s = 8; s >= 1; s >>= 1) {
#pragma unroll
        for (int j = 0; j < KNN; ++j) {
            if ((j & s) == 0) {
                const unsigned long long x = c[j];
                const unsigned long long y = c[j + s];
                const bool sw = y < x;
                c[j]     = sw ? y : x;
                c[j + s] = sw ? x : y;
            }
        }
    }

    // ---- lanes 0..15 own the final lists: apply filters, write edges ----
    if (lane < 16) {
        const int ctr_g   = base + nloc;
        const int ctr_res = res_idx[ctr_g];
        const float4 pctr = sCand[nloc];
#pragma unroll
        for (int r = 0; r < KNN; ++r) {
            const int nbr_loc = (int)(unsigned int)c[r];
            const int nbr_g   = base + nbr_loc;
            int rd = res_idx[nbr_g] - ctr_res;
            rd = rd < 0 ? -rd : rd;
            const float4 pn = sCand[nbr_loc];
            const float dx = pn.x - pctr.x;
            const float dy = pn.y - pctr.y;
            const float dz = pn.z - pctr.z;
            const float dist = sqrtf(dx * dx + dy * dy + dz * dz);
            const bool valid = (rd >= MIN_DIST) && (dist >= EPSF);
            const int o = (ctr_g * KNN + r) * 2;
            out[o + 0] = valid ? nbr_g : -1;
            out[o + 1] = valid ? ctr_g : -1;
        }
    }
}

extern "C" void kernel_launch(void* const* d_in, const int* in_sizes, int n_in,
                              void* d_out, int out_size, void* d_ws, size_t ws_size,
                              hipStream_t stream) {
    (void)in_sizes; (void)n_in; (void)out_size; (void)d_ws; (void)ws_size;
    const float* pos     = (const float*)d_in[0];
    const int*   res_idx = (const int*)d_in[1];
    // d_in[2] (batch) unused: graphs are equal-sized (B x NPG) by construction.
    int* out = (int*)d_out;

    // 16 graphs x 16 blocks; each block: 8 waves x 16 centers = 128 centers.
    knn_wmma_kernel<<<dim3(NGRAPH * 16), dim3(256), 0, stream>>>(pos, res_idx, out);
}